// Attn_5214090297883
// MI455X (gfx1250) — compile-verified
//
#include <hip/hip_runtime.h>
#include <math.h>

typedef __attribute__((ext_vector_type(2))) float v2f;
typedef __attribute__((ext_vector_type(4))) float f32x4;
typedef __attribute__((ext_vector_type(8))) float v8f;

#define S_LEN 2048
#define B_DIM 32
#define H_DIM 1024

// ---------------------------------------------------------------------------
// Kernel 1: u[b][h] = sum_o hidden[b][o] * W[o][h]   (u = hidden @ W)
// GEMM M=B_DIM=32, N=H_DIM=1024, K=H_DIM=1024 using V_WMMA_F32_16X16X4_F32.
// One wave (32 threads) per 16x16 output tile. grid = (N/16, M/16) = (64, 2).
// A fragment (16x4 f32): lanes 0-15 hold M=lane, K={k,k+1}; lanes 16-31 hold
// M=lane-16, K={k+2,k+3}  -> per-lane float2 at hidden[m][k+2*hi].
// B fragment (4x16 f32): lane gives N=lane&15; VGPR0/1 hold rows K=(2*hi)/(2*hi+1).
// D (16x16 f32, 8 VGPRs): VGPR j -> row j + 8*hi, col lane&15.
// ---------------------------------------------------------------------------
__global__ __launch_bounds__(32) void uvec_wmma_kernel(
    const float* __restrict__ hidden,   // [B,H]
    const float* __restrict__ W,        // [H(out), H(in)] row-major
    float* __restrict__ u)              // [B,H]
{
    const int lane  = threadIdx.x;        // 0..31, EXEC all ones
    const int lo    = lane & 15;
    const int hi    = lane >> 4;          // 0 or 1 (K high bit for A/B frags)
    const int nBase = blockIdx.x * 16;    // h tile
    const int mBase = blockIdx.y * 16;    // b tile

    const int m = mBase + lo;             // A row for this lane
    const int n = nBase + lo;             // B/D column for this lane

    v8f acc = {};
    const float* arow = hidden + (size_t)m * H_DIM;

    for (int k = 0; k < H_DIM; k += 4) {
        const int kk = k + 2 * hi;
        // A: two consecutive K values for row m (8-byte aligned: kk is even)
        v2f a = *(const v2f*)(arow + kk);
        // B: W[kk][n], W[kk+1][n]
        v2f b;
        b.x = W[(size_t)kk * H_DIM + n];
        b.y = W[(size_t)(kk + 1) * H_DIM + n];
        acc = __builtin_amdgcn_wmma_f32_16x16x4_f32(
            /*neg_a=*/false, a, /*neg_b=*/false, b,
            /*c_mod=*/(short)0, acc, /*reuse_a=*/false, /*reuse_b=*/false);
    }

#pragma unroll
    for (int j = 0; j < 8; ++j) {
        u[(size_t)(mBase + j + 8 * hi) * H_DIM + n] = acc[j];
    }
}

// ---------------------------------------------------------------------------
// Kernel 2 (bandwidth-dominant): scores[b][s] = enc[s][b][:] . u[b][:]
// grid = (S/8, B), block = 256 (8 waves). u[b] (4KB) staged in LDS once per
// block; each wave owns one s-row: 8 x 128-byte coalesced non-temporal b128
// loads per lane-group, FMA dot, then wave32 shuffle reduction.
// ---------------------------------------------------------------------------
__global__ __launch_bounds__(256) void score_kernel(
    const float* __restrict__ enc,      // [S,B,H]
    const float* __restrict__ u,        // [B,H]
    float* __restrict__ scores)         // [B,S]
{
    __shared__ f32x4 su[H_DIM / 4];     // 4 KB

    const int b   = blockIdx.y;
    const int tid = threadIdx.x;

    // Stage u[b] into LDS: 256 threads x one float4 = 1024 floats.
    su[tid] = ((const f32x4*)(u + (size_t)b * H_DIM))[tid];
    __syncthreads();

    const int lane = tid & 31;
    const int wave = tid >> 5;
    const int s    = blockIdx.x * 8 + wave;

    const f32x4* erow = (const f32x4*)(enc + ((size_t)s * B_DIM + b) * H_DIM);

    float acc = 0.0f;
#pragma unroll
    for (int i = 0; i < 8; ++i) {
        // Streaming 256MB tensor: non-temporal so it bypasses/limits L2 residency.
        f32x4 e  = __builtin_nontemporal_load(&erow[i * 32 + lane]);
        f32x4 uu = su[i * 32 + lane];
        acc = fmaf(e.x, uu.x, acc);
        acc = fmaf(e.y, uu.y, acc);
        acc = fmaf(e.z, uu.z, acc);
        acc = fmaf(e.w, uu.w, acc);
    }

    // wave32 reduction
#pragma unroll
    for (int off = 16; off > 0; off >>= 1)
        acc += __shfl_xor(acc, off, 32);

    if (lane == 0) scores[(size_t)b * S_LEN + s] = acc;
}

// ---------------------------------------------------------------------------
// Kernel 3: out[b,0,s] = softmax_s(scores[b,s]).  grid = B, block = 256,
// 8 elements per thread. (Bias term dropped: softmax shift-invariant.)
// ---------------------------------------------------------------------------
__global__ __launch_bounds__(256) void softmax_kernel(
    const float* __restrict__ scores,   // [B,S]
    float* __restrict__ out)            // [B,1,S]
{
    __shared__ float red[8];

    const int b    = blockIdx.x;
    const int tid  = threadIdx.x;
    const int lane = tid & 31;
    const int wave = tid >> 5;
    const float* row = scores + (size_t)b * S_LEN;

    float v[8];
    float m = -INFINITY;
#pragma unroll
    for (int i = 0; i < 8; ++i) {
        v[i] = row[tid + i * 256];
        m = fmaxf(m, v[i]);
    }
#pragma unroll
    for (int off = 16; off > 0; off >>= 1)
        m = fmaxf(m, __shfl_xor(m, off, 32));
    if (lane == 0) red[wave] = m;
    __syncthreads();
    float bm = red[0];
#pragma unroll
    for (int w = 1; w < 8; ++w) bm = fmaxf(bm, red[w]);
    __syncthreads();

    float sum = 0.0f;
#pragma unroll
    for (int i = 0; i < 8; ++i) {
        v[i] = expf(v[i] - bm);
        sum += v[i];
    }
#pragma unroll
    for (int off = 16; off > 0; off >>= 1)
        sum += __shfl_xor(sum, off, 32);
    if (lane == 0) red[wave] = sum;
    __syncthreads();
    float bs = 0.0f;
#pragma unroll
    for (int w = 0; w < 8; ++w) bs += red[w];
    const float inv = 1.0f / bs;

#pragma unroll
    for (int i = 0; i < 8; ++i)
        out[(size_t)b * S_LEN + tid + i * 256] = v[i] * inv;
}

extern "C" void kernel_launch(void* const* d_in, const int* in_sizes, int n_in,
                              void* d_out, int out_size, void* d_ws, size_t ws_size,
                              hipStream_t stream) {
    (void)in_sizes; (void)n_in; (void)out_size; (void)ws_size;

    const float* hidden = (const float*)d_in[0];  // [B,H]
    const float* enc    = (const float*)d_in[1];  // [S,B,H]
    const float* W      = (const float*)d_in[2];  // [H,H]
    // d_in[3] = bias: provably irrelevant under softmax (shift invariance).

    float* out = (float*)d_out;                   // [B,1,S]

    float* ws     = (float*)d_ws;
    float* u      = ws;                           // B*H floats   (128 KB)
    float* scores = ws + (size_t)B_DIM * H_DIM;   // B*S floats   (256 KB)

    // 1) u = hidden @ W via V_WMMA_F32_16X16X4_F32 (one wave per 16x16 tile)
    uvec_wmma_kernel<<<dim3(H_DIM / 16, B_DIM / 16), 32, 0, stream>>>(hidden, W, u);

    // 2) streaming dot: scores[b][s] = enc[s,b,:] . u[b,:]
    score_kernel<<<dim3(S_LEN / 8, B_DIM), 256, 0, stream>>>(enc, u, scores);

    // 3) row softmax -> out
    softmax_kernel<<<B_DIM, 256, 0, stream>>>(scores, out);
}